// PriorLSTM_17016660426800
// MI455X (gfx1250) — compile-verified
//
#include <hip/hip_runtime.h>
#include <math.h>

// Problem constants (match reference)
#define B_    16
#define T_    32
#define BT_   512          // B*T
#define IN_   52686
#define H_    128
#define G4_   512          // 4*H
#define L_    16
#define OUT_  2976
#define EPSN  1e-5f
#define BK    64           // K-chunk staged in LDS for GEMMs
#define LP    (BK + 2)     // LDS row stride (keeps rows 8B aligned, no bank conflicts)
#define WS_   (H_ + 2)     // LDS row stride for h / Whh in the layer kernel

typedef float v2f __attribute__((ext_vector_type(2)));
typedef float v8f __attribute__((ext_vector_type(8)));

__device__ __forceinline__ v8f wmma4_f32(v2f a, v2f b, v8f c) {
  // V_WMMA_F32_16X16X4_F32: D = A(16x4) * B(4x16) + C, all fp32
  return __builtin_amdgcn_wmma_f32_16x16x4_f32(false, a, false, b, (short)0, c,
                                               false, false);
}

__device__ __forceinline__ float sigm(float x) { return 1.0f / (1.0f + expf(-x)); }

// ---------------------------------------------------------------------------
// K1: BatchNorm stats folded into per-channel scale a[d] and shift c[d]:
//     xn[bt,d] = x[bt,d]*a[d] + c[d]
// ---------------------------------------------------------------------------
__global__ __launch_bounds__(256) void bn_stats(const float* __restrict__ x,
                                                const float* __restrict__ gamma,
                                                const float* __restrict__ beta,
                                                float* __restrict__ a,
                                                float* __restrict__ c) {
  int d = blockIdx.x * 256 + threadIdx.x;
  if (d >= IN_) return;
  float s = 0.f, s2 = 0.f;
  for (int bt = 0; bt < BT_; ++bt) {
    float v = x[(long)bt * IN_ + d];
    s += v; s2 += v * v;
  }
  float mean = s * (1.0f / BT_);
  float var  = s2 * (1.0f / BT_) - mean * mean;
  float av   = gamma[d] * rsqrtf(var + EPSN);
  a[d] = av;
  c[d] = beta[d] - mean * av;
}

// ---------------------------------------------------------------------------
// K2: big front GEMM  GX[m,g] = sum_d ( A_virt[m,d] * W0[g,d] )
//     A_virt rows 0..511 = x*a (BN scale fused), row 512 = c (BN shift),
//     rows 513..527 = 0.  One wave per 16x16 tile, f32 WMMA, float2 loads.
// ---------------------------------------------------------------------------
__global__ __launch_bounds__(32) void gemm_layer0(const float* __restrict__ x,
                                                  const float* __restrict__ a,
                                                  const float* __restrict__ c,
                                                  const float* __restrict__ W0,
                                                  float* __restrict__ GX) {
  __shared__ float As[16][LP];
  __shared__ float Bs[16][LP];
  const int lane = threadIdx.x;          // 0..31
  const int nt = blockIdx.x;             // 0..31  (gate tiles)
  const int mt = blockIdx.y;             // 0..32  (row tiles, incl. virtual)
  const int m0 = mt * 16, n0 = nt * 16;
  const int nn = lane & 15, hl = lane >> 4;
  v8f acc = {};

  for (int k0 = 0; k0 < IN_; k0 += BK) {
    if (k0 + BK <= IN_) {                // full chunk: vectorized b64 path
      #pragma unroll 4
      for (int r = 0; r < 16; ++r) {     // row r, lane -> float2 column
        int k = k0 + lane * 2;
        int gr = m0 + r;
        float2 av = make_float2(0.f, 0.f);
        if (gr < BT_) {
          float2 xv = *(const float2*)&x[(long)gr * IN_ + k];
          float2 aa = *(const float2*)&a[k];
          av.x = xv.x * aa.x; av.y = xv.y * aa.y;
        } else if (gr == BT_) {
          av = *(const float2*)&c[k];
        }
        float2 bv = *(const float2*)&W0[(long)(n0 + r) * IN_ + k];
        *(float2*)&As[r][lane * 2] = av;
        *(float2*)&Bs[r][lane * 2] = bv;
        if (k0 + BK + BK <= IN_) {       // prefetch next chunk (global_prefetch_b8)
          if (gr < BT_) __builtin_prefetch(&x[(long)gr * IN_ + k0 + BK], 0, 3);
          __builtin_prefetch(&W0[(long)(n0 + r) * IN_ + k0 + BK], 0, 3);
        }
      }
    } else {                             // tail chunk: scalar, zero padded
      for (int i = 0; i < 32; ++i) {
        int idx = i * 32 + lane;
        int r = idx >> 6, col = idx & 63;
        int k = k0 + col;
        float av = 0.f, bv = 0.f;
        if (k < IN_) {
          int gr = m0 + r;
          if (gr < BT_)        av = x[(long)gr * IN_ + k] * a[k];
          else if (gr == BT_)  av = c[k];
          bv = W0[(long)(n0 + r) * IN_ + k];
        }
        As[r][col] = av;
        Bs[r][col] = bv;
      }
    }
    __syncthreads();
    #pragma unroll
    for (int kk = 0; kk < BK; kk += 4) {
      int kb = kk + 2 * hl;              // even -> 8B-aligned LDS b64 loads
      v2f a2 = *(const v2f*)&As[nn][kb];
      v2f b2 = *(const v2f*)&Bs[nn][kb];
      acc = wmma4_f32(a2, b2, acc);
    }
    __syncthreads();
  }
  #pragma unroll
  for (int r = 0; r < 8; ++r) {
    int m = m0 + r + 8 * hl;
    GX[(long)m * G4_ + n0 + nn] = acc[r];
  }
}

// K3: add virtual shift row + layer-0 biases to all 512 real rows
__global__ __launch_bounds__(256) void fixup0(float* __restrict__ GX,
                                              const float* __restrict__ b_ih0,
                                              const float* __restrict__ b_hh0) {
  int i = blockIdx.x * 256 + threadIdx.x;
  if (i >= BT_ * G4_) return;
  int g = i & (G4_ - 1);
  GX[i] += GX[BT_ * G4_ + g] + b_ih0[g] + b_hh0[g];
}

// ---------------------------------------------------------------------------
// Generic NT WMMA GEMM: D[m,n] = sum_k A[m,k]*Bw[n,k] + bias1[n] + bias2[n]
// One wave per 16x16 tile. M,N multiples of 16; lda/ldb even; K arbitrary.
// ---------------------------------------------------------------------------
__global__ __launch_bounds__(32) void wmma_gemm_nt(const float* __restrict__ A, int lda,
                                                   const float* __restrict__ Bw, int ldb,
                                                   const float* __restrict__ bias1,
                                                   const float* __restrict__ bias2,
                                                   float* __restrict__ D, int ldd, int K) {
  __shared__ float As[16][LP];
  __shared__ float Bs[16][LP];
  const int lane = threadIdx.x;
  const int nt = blockIdx.x, mt = blockIdx.y;
  const int m0 = mt * 16, n0 = nt * 16;
  const int nn = lane & 15, hl = lane >> 4;
  v8f acc = {};

  for (int k0 = 0; k0 < K; k0 += BK) {
    if (k0 + BK <= K) {
      #pragma unroll 4
      for (int r = 0; r < 16; ++r) {
        int k = k0 + lane * 2;
        *(float2*)&As[r][lane * 2] = *(const float2*)&A[(long)(m0 + r) * lda + k];
        *(float2*)&Bs[r][lane * 2] = *(const float2*)&Bw[(long)(n0 + r) * ldb + k];
      }
    } else {
      for (int i = 0; i < 32; ++i) {
        int idx = i * 32 + lane;
        int r = idx >> 6, col = idx & 63;
        int k = k0 + col;
        As[r][col] = (k < K) ? A[(long)(m0 + r) * lda + k] : 0.f;
        Bs[r][col] = (k < K) ? Bw[(long)(n0 + r) * ldb + k] : 0.f;
      }
    }
    __syncthreads();
    #pragma unroll
    for (int kk = 0; kk < BK; kk += 4) {
      int kb = kk + 2 * hl;
      v2f a2 = *(const v2f*)&As[nn][kb];
      v2f b2 = *(const v2f*)&Bs[nn][kb];
      acc = wmma4_f32(a2, b2, acc);
    }
    __syncthreads();
  }
  float bv = 0.f;
  if (bias1) bv += bias1[n0 + nn];
  if (bias2) bv += bias2[n0 + nn];
  #pragma unroll
  for (int r = 0; r < 8; ++r) {
    int m = m0 + r + 8 * hl;
    D[(long)m * ldd + n0 + nn] = acc[r] + bv;
  }
}

// ---------------------------------------------------------------------------
// K5: one ENTIRE LSTM layer (all 32 timesteps) in a single persistent block.
// CDNA5 320KB LDS: Whh (512x128 f32, 256KB) staged once; h lives in LDS;
// c stays in registers (each lane owns the same (batch,hidden) pairs every
// step because the WMMA C-layout is time-invariant). 8 waves; wave w owns
// hidden slice j in [16w,16w+16) and computes all 4 gate tiles with WMMA
// (M=batch=16, N=16, K=128), then gates fully in registers.
// ---------------------------------------------------------------------------
__global__ __launch_bounds__(256) void lstm_layer(const float* __restrict__ GX,
                                                  const float* __restrict__ Whh,
                                                  float* __restrict__ Hout) {
  extern __shared__ float smem[];
  float* Whs = smem;                  // [512][WS_]  (256KB + pad)
  float* Hs  = smem + G4_ * WS_;      // [16][WS_]   h state, Hs[b*WS_ + j]

  // Stage Whh into LDS with padded rows (b64 copies)
  for (int i = threadIdx.x; i < G4_ * (H_ / 2); i += 256) {
    int g = i >> 6, c2 = i & 63;
    *(float2*)&Whs[g * WS_ + c2 * 2] = *(const float2*)&Whh[g * H_ + c2 * 2];
  }
  __syncthreads();

  const int lane = threadIdx.x & 31;
  const int wv   = threadIdx.x >> 5;       // 0..7 -> hidden slice
  const int nn = lane & 15, hl = lane >> 4;
  const int j = wv * 16 + nn;              // hidden index owned by this lane
  const int ri = (0 * H_ + j) * WS_;       // Whh LDS row offsets per gate
  const int rf = (1 * H_ + j) * WS_;
  const int rg = (2 * H_ + j) * WS_;
  const int ro = (3 * H_ + j) * WS_;

  v8f cacc = {};                           // c state, resident in VGPRs

  for (int t = 0; t < T_; ++t) {
    v8f ai = {}, af = {}, ag = {}, ao = {};
    if (t > 0) {
      #pragma unroll
      for (int kk = 0; kk < H_; kk += 4) {
        int kb = kk + 2 * hl;              // even -> b64 LDS loads
        v2f a2 = *(const v2f*)&Hs[nn * WS_ + kb];
        v2f bi = *(const v2f*)&Whs[ri + kb];
        v2f bf = *(const v2f*)&Whs[rf + kb];
        v2f bg = *(const v2f*)&Whs[rg + kb];
        v2f bo = *(const v2f*)&Whs[ro + kb];
        ai = wmma4_f32(a2, bi, ai);
        af = wmma4_f32(a2, bf, af);
        ag = wmma4_f32(a2, bg, ag);
        ao = wmma4_f32(a2, bo, ao);
      }
    }
    float hn[8];
    #pragma unroll
    for (int r = 0; r < 8; ++r) {
      int b = r + 8 * hl;                  // batch index
      long row = (long)(b * T_ + t) * G4_;
      float gi = ai[r] + GX[row + 0 * H_ + j];
      float gf = af[r] + GX[row + 1 * H_ + j];
      float gg = ag[r] + GX[row + 2 * H_ + j];
      float go = ao[r] + GX[row + 3 * H_ + j];
      float cn = sigm(gf) * cacc[r] + sigm(gi) * tanhf(gg);
      cacc[r] = cn;
      hn[r] = sigm(go) * tanhf(cn);
    }
    __syncthreads();                       // all reads of old h done
    #pragma unroll
    for (int r = 0; r < 8; ++r) {
      int b = r + 8 * hl;
      Hs[b * WS_ + j] = hn[r];
      Hout[(long)(b * T_ + t) * H_ + j] = hn[r];
    }
    __syncthreads();                       // new h visible to all waves
  }
}

// ---------------------------------------------------------------------------
// K6: temporal max-pool + prior logits (b_out already added in GEMM)
// ---------------------------------------------------------------------------
__global__ __launch_bounds__(256) void maxpool_prior(const float* __restrict__ OutFull,
                                                     const float* __restrict__ prior,
                                                     float* __restrict__ out) {
  int i = blockIdx.x * 256 + threadIdx.x;
  if (i >= B_ * OUT_) return;
  int b = i / OUT_, o = i - b * OUT_;
  float m = -3.402823466e+38f;
  for (int t = 0; t < T_; ++t)
    m = fmaxf(m, OutFull[(long)(b * T_ + t) * OUT_ + o]);
  float p = 0.f;
  if (o > 0) {
    float pv = prior[o - 1];
    pv = fminf(fmaxf(pv, 1e-8f), 1.f - 1e-8f);
    p = logf(pv) - log1pf(-pv);
  }
  out[i] = m + p;
}

// ---------------------------------------------------------------------------
extern "C" void kernel_launch(void* const* d_in, const int* in_sizes, int n_in,
                              void* d_out, int out_size, void* d_ws, size_t ws_size,
                              hipStream_t stream) {
  const float* x     = (const float*)d_in[0];
  const float* gamma = (const float*)d_in[1];
  const float* beta  = (const float*)d_in[2];
  const float* W_ih0 = (const float*)d_in[3];
  const float* W_ih  = (const float*)d_in[4];   // [15,512,128]
  const float* W_hh  = (const float*)d_in[5];   // [16,512,128]
  const float* b_ih  = (const float*)d_in[6];   // [16,512]
  const float* b_hh  = (const float*)d_in[7];   // [16,512]
  const float* W_out = (const float*)d_in[8];   // [2976,128]
  const float* b_out = (const float*)d_in[9];
  const float* prior = (const float*)d_in[10];
  float* out = (float*)d_out;

  // Workspace layout (floats)
  float* ws = (float*)d_ws;
  float* a_s   = ws;                         // IN_
  float* c_s   = ws + IN_;                   // IN_
  float* GX    = ws + 105472;                // 528*512 (33 M-tiles incl. virtual)
  float* Hbuf0 = ws + 375808;                // 512*128
  float* Hbuf1 = ws + 441344;                // 512*128
  float* OutF  = ws + 510976;                // 512*2976

  const size_t layer_smem = (size_t)(G4_ * WS_ + 16 * WS_) * sizeof(float); // ~270KB

  // 1) BN -> scale/shift
  bn_stats<<<(IN_ + 255) / 256, 256, 0, stream>>>(x, gamma, beta, a_s, c_s);

  // 2) big input GEMM with fused BN (virtual shift row = M-tile 32)
  gemm_layer0<<<dim3(G4_ / 16, 33), 32, 0, stream>>>(x, a_s, c_s, W_ih0, GX);
  fixup0<<<(BT_ * G4_ + 255) / 256, 256, 0, stream>>>(GX, b_ih, b_hh);

  // 3) layer 0 recurrence (whole layer in one persistent kernel)
  lstm_layer<<<1, 256, layer_smem, stream>>>(GX, W_hh, Hbuf0);

  // 4) layers 1..15
  for (int l = 1; l < L_; ++l) {
    float* Hprev = (l & 1) ? Hbuf0 : Hbuf1;
    float* Hcur  = (l & 1) ? Hbuf1 : Hbuf0;
    wmma_gemm_nt<<<dim3(G4_ / 16, BT_ / 16), 32, 0, stream>>>(
        Hprev, H_, W_ih + (long)(l - 1) * G4_ * H_, H_,
        b_ih + l * G4_, b_hh + l * G4_, GX, G4_, H_);
    lstm_layer<<<1, 256, layer_smem, stream>>>(GX, W_hh + (long)l * G4_ * H_, Hcur);
  }

  // 5) output projection (+b_out), then temporal max + prior
  wmma_gemm_nt<<<dim3(OUT_ / 16, BT_ / 16), 32, 0, stream>>>(
      Hbuf1, H_, W_out, H_, b_out, nullptr, OutF, OUT_, H_);
  maxpool_prior<<<(B_ * OUT_ + 255) / 256, 256, 0, stream>>>(OutF, prior, out);
}